// GNNLayer_62526133895859
// MI455X (gfx1250) — compile-verified
//
#include <hip/hip_runtime.h>
#include <math.h>

#define IN_C  128
#define OUT_C 128
#define LDSW  132   // padded LDS row stride (floats) -> conflict-free A-fragment reads

typedef float v2f __attribute__((ext_vector_type(2)));
typedef float v8f __attribute__((ext_vector_type(8)));

// ---------------------------------------------------------------------------
// Kernel 1: edge scatter. One wave (32 lanes) per edge; each lane moves 4
// channels (float4 load from x[col], 4x global_atomic_add_f32 into nsum[row]).
// Lane 0 also bumps the degree counter.
// ---------------------------------------------------------------------------
__global__ __launch_bounds__(256) void gnn_scatter(const float* __restrict__ x,
                                                   const int* __restrict__ ei,
                                                   float* __restrict__ nsum,
                                                   float* __restrict__ deg,
                                                   int E) {
    int e = blockIdx.x * 8 + (threadIdx.x >> 5);
    if (e >= E) return;
    int lane = threadIdx.x & 31;
    int row = ei[e];        // destination
    int col = ei[E + e];    // source / neighbor
    const float4 v = *(const float4*)(x + (size_t)col * IN_C + lane * 4);
    float* dst = nsum + (size_t)row * IN_C + lane * 4;
    atomicAdd(dst + 0, v.x);
    atomicAdd(dst + 1, v.y);
    atomicAdd(dst + 2, v.z);
    atomicAdd(dst + 3, v.w);
    if (lane == 0) atomicAdd(deg + row, 1.0f);
}

// ---------------------------------------------------------------------------
// Kernel 2: fused dual GEMM with V_WMMA_F32_16X16X4_F32.
// Block = 256 threads = 8 waves. Block handles 16 rows; wave w handles the
// 16-wide output column tile n0 = 16*w. A-tiles (x and neighbor_mean) staged
// in LDS (coalesced, padded rows). B fragments read from W rows in global
// (L2-resident, reused by all blocks). out = A_x*Wself^T + A_nm*Wnb^T + bias.
// ---------------------------------------------------------------------------
__global__ __launch_bounds__(256) void gnn_gemm(const float* __restrict__ x,
                                                const float* __restrict__ nsum,
                                                const float* __restrict__ deg,
                                                const float* __restrict__ Wself,
                                                const float* __restrict__ bself,
                                                const float* __restrict__ Wnb,
                                                const float* __restrict__ bnb,
                                                float* __restrict__ out) {
    __shared__ float lx[16 * LDSW];
    __shared__ float ln[16 * LDSW];
    __shared__ float linv[16];

    const int tid  = threadIdx.x;
    const int row0 = blockIdx.x * 16;

    if (tid < 16) {
        float d = deg[row0 + tid];
        linv[tid] = 1.0f / fmaxf(d, 1.0f);
    }
    const float* xsrc = x + (size_t)row0 * IN_C;
    for (int idx = tid; idx < 16 * IN_C; idx += 256)
        lx[(idx >> 7) * LDSW + (idx & 127)] = xsrc[idx];
    __syncthreads();
    const float* nsrc = nsum + (size_t)row0 * IN_C;
    for (int idx = tid; idx < 16 * IN_C; idx += 256)
        ln[(idx >> 7) * LDSW + (idx & 127)] = nsrc[idx] * linv[idx >> 7];
    __syncthreads();

    const int wave = tid >> 5;          // 0..7 -> output column tile
    const int lane = tid & 31;
    const int m    = lane & 15;         // A row / B,C,D column within tile
    const int kb   = (lane >> 4) << 1;  // K sub-offset: 0 or 2
    const int n0   = wave * 16;

    // B[k][n] = W[n0+n][k] : W rows, 2 consecutive floats per lane per step
    const float* wsrow = Wself + (size_t)(n0 + m) * IN_C;
    const float* wnrow = Wnb   + (size_t)(n0 + m) * IN_C;
    const float* lxr = lx + m * LDSW;
    const float* lnr = ln + m * LDSW;

    v8f c = {0.f, 0.f, 0.f, 0.f, 0.f, 0.f, 0.f, 0.f};
#pragma unroll 4
    for (int k = 0; k < IN_C; k += 4) {
        v2f a, b;
        // self path: A = x tile
        a.x = lxr[k + kb];
        a.y = lxr[k + kb + 1];
        b.x = wsrow[k + kb];
        b.y = wsrow[k + kb + 1];
        c = __builtin_amdgcn_wmma_f32_16x16x4_f32(false, a, false, b,
                                                  (short)0, c, false, false);
        // neighbor path: A = neighbor_mean tile
        a.x = lnr[k + kb];
        a.y = lnr[k + kb + 1];
        b.x = wnrow[k + kb];
        b.y = wnrow[k + kb + 1];
        c = __builtin_amdgcn_wmma_f32_16x16x4_f32(false, a, false, b,
                                                  (short)0, c, false, false);
    }

    const float bias = bself[n0 + m] + bnb[n0 + m];
    const int hi = lane >> 4;  // lanes 16-31 hold M = v+8
    float* orow = out + (size_t)(row0 + 8 * hi) * OUT_C + n0 + m;
#pragma unroll
    for (int v = 0; v < 8; ++v)
        orow[(size_t)v * OUT_C] = c[v] + bias;
}

// ---------------------------------------------------------------------------
// Kernel 3: per-channel sum / sum-of-squares (privatized per block, one
// atomicAdd pair per block per channel). 128 threads = 1 channel each.
// ---------------------------------------------------------------------------
__global__ __launch_bounds__(128) void gnn_stats(const float* __restrict__ out,
                                                 float* __restrict__ stats,
                                                 int N) {
    int c = threadIdx.x;
    float s = 0.f, q = 0.f;
    for (int r = blockIdx.x; r < N; r += gridDim.x) {
        float v = out[(size_t)r * OUT_C + c];
        s += v;
        q += v * v;
    }
    atomicAdd(stats + c, s);
    atomicAdd(stats + 128 + c, q);
}

// ---------------------------------------------------------------------------
// Kernel 4: fold batch stats + gamma/beta into per-channel scale & shift.
// ---------------------------------------------------------------------------
__global__ __launch_bounds__(128) void gnn_bnparam(float* __restrict__ stats,
                                                   const float* __restrict__ gamma,
                                                   const float* __restrict__ beta,
                                                   int N) {
    int c = threadIdx.x;
    float invN = 1.0f / (float)N;
    float mean = stats[c] * invN;
    float var  = stats[128 + c] * invN - mean * mean;
    float sc   = gamma[c] * rsqrtf(var + 1e-5f);
    stats[256 + c] = sc;
    stats[384 + c] = beta[c] - mean * sc;
}

// ---------------------------------------------------------------------------
// Kernel 5: in-place normalize + ReLU, vectorized float4.
// ---------------------------------------------------------------------------
__global__ __launch_bounds__(256) void gnn_final(float* __restrict__ out,
                                                 const float* __restrict__ stats,
                                                 long long total4) {
    long long i = (long long)blockIdx.x * blockDim.x + threadIdx.x;
    if (i >= total4) return;
    int c4 = (int)(i & 31) * 4;   // channel group (OUT_C/4 == 32)
    const float4 sc = *(const float4*)(stats + 256 + c4);
    const float4 sh = *(const float4*)(stats + 384 + c4);
    float4 v = ((float4*)out)[i];
    v.x = fmaxf(fmaf(v.x, sc.x, sh.x), 0.f);
    v.y = fmaxf(fmaf(v.y, sc.y, sh.y), 0.f);
    v.z = fmaxf(fmaf(v.z, sc.z, sh.z), 0.f);
    v.w = fmaxf(fmaf(v.w, sc.w, sh.w), 0.f);
    ((float4*)out)[i] = v;
}

// ---------------------------------------------------------------------------
extern "C" void kernel_launch(void* const* d_in, const int* in_sizes, int n_in,
                              void* d_out, int out_size, void* d_ws, size_t ws_size,
                              hipStream_t stream) {
    const float* x     = (const float*)d_in[0];
    const int*   ei    = (const int*)d_in[1];
    const float* Wself = (const float*)d_in[2];
    const float* bself = (const float*)d_in[3];
    const float* Wnb   = (const float*)d_in[4];
    const float* bnb   = (const float*)d_in[5];
    const float* gamma = (const float*)d_in[6];
    const float* beta  = (const float*)d_in[7];
    float* out = (float*)d_out;

    const int N = in_sizes[0] / IN_C;   // 100000
    const int E = in_sizes[1] / 2;      // 1600000

    // workspace layout: [nsum N*128][deg N][stats 512]
    float* nsum  = (float*)d_ws;
    float* deg   = nsum + (size_t)N * IN_C;
    float* stats = deg + N;
    size_t zero_bytes = ((size_t)N * IN_C + (size_t)N + 512) * sizeof(float);
    hipMemsetAsync(d_ws, 0, zero_bytes, stream);

    // 1) scatter-sum neighbors + degree (8 edges per 256-thread block)
    int eblocks = (E + 7) / 8;
    gnn_scatter<<<eblocks, 256, 0, stream>>>(x, ei, nsum, deg, E);

    // 2) fused dual WMMA GEMM + bias -> pre-BN activations in d_out
    gnn_gemm<<<N / 16, 256, 0, stream>>>(x, nsum, deg, Wself, bself, Wnb, bnb, out);

    // 3) batch statistics
    gnn_stats<<<512, 128, 0, stream>>>(out, stats, N);

    // 4) fold into scale/shift
    gnn_bnparam<<<1, 128, 0, stream>>>(stats, gamma, beta, N);

    // 5) normalize + ReLU in place
    long long total4 = (long long)N * (OUT_C / 4);
    int fblocks = (int)((total4 + 255) / 256);
    gnn_final<<<fblocks, 256, 0, stream>>>(out, stats, total4);
}